// GATv3_65687229825025
// MI455X (gfx1250) — compile-verified
//
#include <hip/hip_runtime.h>
#include <hip/hip_bf16.h>
#include <stdint.h>

typedef float v2f __attribute__((ext_vector_type(2)));
typedef float v8f __attribute__((ext_vector_type(8)));

#define NEG_SLOPE 0.2f
#define BLK 16          // subspace block width (matches WMMA 16x16 tile)
#define POWER_ITERS 24  // subspace iterations
#define GRAM_BLOCKS 64  // waves producing partial Gram matrices

// ---------------- utility device functions ----------------

__device__ __forceinline__ float hash11(uint32_t x) {
  x ^= x >> 16; x *= 0x7feb352dU; x ^= x >> 15; x *= 0x846ca68bU; x ^= x >> 16;
  return (float)(x & 0xFFFFFFu) * (1.0f / 8388608.0f) - 1.0f;   // [-1,1)
}

// order-preserving float->uint key for atomicMax-based segment max
__device__ __forceinline__ unsigned fkey(float f) {
  unsigned u = __float_as_uint(f);
  return u ^ ((u & 0x80000000u) ? 0xFFFFFFFFu : 0x80000000u);
}
__device__ __forceinline__ float fdec(unsigned k) {
  unsigned u = (k & 0x80000000u) ? (k ^ 0x80000000u) : ~k;
  return __uint_as_float(u);
}

// ---------------- degree / CSR construction (one-time) ----------------

__global__ void k_zero(float* p, int n) {
  for (int i = blockIdx.x * blockDim.x + threadIdx.x; i < n; i += gridDim.x * blockDim.x)
    p[i] = 0.0f;
}

__global__ void k_deg(const int* __restrict__ src, float* __restrict__ deg, int E) {
  int i = blockIdx.x * blockDim.x + threadIdx.x;
  if (i < E) atomicAdd(&deg[src[i]], 1.0f);
}

__global__ void k_dinv(const float* __restrict__ deg, float* __restrict__ dinv,
                       float* __restrict__ mfl, int N) {
  int i = blockIdx.x * blockDim.x + threadIdx.x;
  if (i < N) {
    float d = deg[i];
    bool nz = d > 0.0f;
    mfl[i] = nz ? 1.0f : 0.0f;
    dinv[i] = nz ? rsqrtf(d) : 0.0f;
  }
}

// exclusive prefix over integer degrees (one-time, 4096 adds on one thread)
__global__ void k_scan(const float* __restrict__ deg, int* __restrict__ rowstart,
                       int* __restrict__ cursor, int N) {
  if (threadIdx.x || blockIdx.x) return;
  int acc = 0;
  for (int n = 0; n < N; ++n) {
    rowstart[n] = acc;
    cursor[n] = 0;
    acc += (int)deg[n];
  }
  rowstart[N] = acc;
}

__global__ void k_fill(const int* __restrict__ src, const int* __restrict__ dst,
                       const int* __restrict__ rowstart, int* __restrict__ cursor,
                       int* __restrict__ csr_dst, int E) {
  int e = blockIdx.x * blockDim.x + threadIdx.x;
  if (e >= E) return;
  int s = src[e];
  int p = atomicAdd(&cursor[s], 1);
  csr_dst[rowstart[s] + p] = dst[e];
}

__global__ void k_vinit(float* __restrict__ V, int n16) {
  for (int i = blockIdx.x * blockDim.x + threadIdx.x; i < n16; i += gridDim.x * blockDim.x)
    V[i] = hash11((uint32_t)i * 2654435761u + 12345u);
}

// ---------------- gather SpMV:  W = (2I - L) V  (no atomics) ----------------
// L = diag(m) - Dinv A Dinv  =>  W[n] = (2 - m[n]) V[n] + dinv[n] * sum_d dinv[d] V[d]

__global__ void k_spmv(const int* __restrict__ rowstart, const int* __restrict__ csr_dst,
                       const float* __restrict__ dinv, const float* __restrict__ mfl,
                       const float* __restrict__ V, float* __restrict__ W, int N) {
  int i = blockIdx.x * blockDim.x + threadIdx.x;
  if (i >= N * BLK) return;
  int n = i >> 4;
  int c = i & 15;
  float acc = (2.0f - mfl[n]) * V[i];
  float coef = dinv[n];
  int r1 = rowstart[n + 1];
  for (int e = rowstart[n]; e < r1; ++e) {
    int d = csr_dst[e];
    acc += coef * dinv[d] * V[d * BLK + c];
  }
  W[i] = acc;
}

// ---------------- WMMA kernels (V_WMMA_F32_16X16X4_F32) ----------------
// Gpart[wave][m][n] = partial of sum_node P[node][m] * Q[node][n]
// A[m][k] = P[n0+k][m], B[k][n] = Q[n0+k][n]; K=4 node chunk per WMMA.
// 2x unrolled with independent accumulators for memory-level parallelism.

__global__ __launch_bounds__(32) void k_gram(const float* __restrict__ P,
                                             const float* __restrict__ Q,
                                             float* __restrict__ Gpart, int N) {
  int lane = threadIdx.x;
  int half = lane >> 4;      // 0: K0/K1, 1: K2/K3
  int l = lane & 15;
  v8f acc0 = {}, acc1 = {};
  int ngroups = N >> 2;      // 4 nodes per WMMA
  for (int g = 2 * blockIdx.x; g < ngroups; g += 2 * gridDim.x) {
    int base0 = ((g << 2) + 2 * half) * BLK + l;
    v2f a0; a0.x = P[base0]; a0.y = P[base0 + BLK];
    v2f b0; b0.x = Q[base0]; b0.y = Q[base0 + BLK];
    acc0 = __builtin_amdgcn_wmma_f32_16x16x4_f32(false, a0, false, b0,
                                                 (short)0, acc0, false, false);
    if (g + 1 < ngroups) {
      int base1 = base0 + 4 * BLK;
      v2f a1; a1.x = P[base1]; a1.y = P[base1 + BLK];
      v2f b1; b1.x = Q[base1]; b1.y = Q[base1 + BLK];
      acc1 = __builtin_amdgcn_wmma_f32_16x16x4_f32(false, a1, false, b1,
                                                   (short)0, acc1, false, false);
    }
  }
  acc0 = acc0 + acc1;
  float* Gout = Gpart + (size_t)blockIdx.x * 256;
#pragma unroll
  for (int j = 0; j < 8; ++j)
    Gout[(j + 8 * half) * BLK + l] = acc0[j];   // C/D layout: VGPR j -> row j/j+8
}

// V = W * C   (W: N x 16 row-major, C: 16 x 16 row-major)
__global__ __launch_bounds__(32) void k_apply(const float* __restrict__ W,
                                              const float* __restrict__ C,
                                              float* __restrict__ V, int N) {
  __shared__ float Cs[256];
  int lane = threadIdx.x;
  for (int i = lane; i < 256; i += 32) Cs[i] = C[i];
  __syncthreads();
  int half = lane >> 4;
  int l = lane & 15;
  int ntiles = N >> 4;       // 16 nodes per tile
  for (int tile = blockIdx.x; tile < ntiles; tile += gridDim.x) {
    int n0 = tile << 4;
    v8f acc = {};
#pragma unroll
    for (int kc = 0; kc < 16; kc += 4) {
      int addr = (n0 + l) * BLK + kc + 2 * half;       // A: M=node-in-tile, K=col chunk
      v2f a; a.x = W[addr]; a.y = W[addr + 1];
      v2f b; b.x = Cs[(kc + 2 * half) * BLK + l];      // B rows K0|K2 , K1|K3
      b.y = Cs[(kc + 2 * half + 1) * BLK + l];
      acc = __builtin_amdgcn_wmma_f32_16x16x4_f32(false, a, false, b,
                                                  (short)0, acc, false, false);
    }
#pragma unroll
    for (int j = 0; j < 8; ++j)
      V[(n0 + j + 8 * half) * BLK + l] = acc[j];
  }
}

// ---------------- fused reduce + small dense solves ----------------

// reduce partial Grams in LDS, then C = (L^T)^{-1} with G = L L^T
__global__ __launch_bounds__(256) void k_cholinv(const float* __restrict__ Gpart,
                                                 float* __restrict__ C) {
  __shared__ float Gs[256];
  int tid = threadIdx.x;
  float s = 0.0f;
  for (int w = 0; w < GRAM_BLOCKS; ++w) s += Gpart[(size_t)w * 256 + tid];
  Gs[tid] = s;
  __syncthreads();
  if (tid != 0) return;
  float L[BLK][BLK];
  for (int i = 0; i < BLK; ++i)
    for (int j = 0; j < BLK; ++j) L[i][j] = 0.0f;
  for (int i = 0; i < BLK; ++i) {
    for (int j = 0; j <= i; ++j) {
      float v = Gs[i * BLK + j];
      if (i == j) v += 1e-6f + 1e-6f * fabsf(v);       // ridge for rank safety
      for (int k = 0; k < j; ++k) v -= L[i][k] * L[j][k];
      if (i == j) L[i][i] = sqrtf(fmaxf(v, 1e-20f));
      else        L[i][j] = v / L[j][j];
    }
  }
  // back-substitution: U = L^T (upper), solve U * C = I  (C upper triangular)
  for (int i = 0; i < BLK * BLK; ++i) C[i] = 0.0f;
  for (int j = 0; j < BLK; ++j) {
    C[j * BLK + j] = 1.0f / L[j][j];
    for (int i = j - 1; i >= 0; --i) {
      float v = 0.0f;
      for (int k = i + 1; k <= j; ++k) v += L[k][i] * C[k * BLK + j];  // U[i][k]=L[k][i]
      C[i * BLK + j] = -v / L[i][i];
    }
  }
}

// reduce partial T = V^T B V, then top-2 Ritz vectors; emit #2
__global__ __launch_bounds__(256) void k_smalleig(const float* __restrict__ Tpart,
                                                  float* __restrict__ wv) {
  __shared__ float Ts[256];
  int tid = threadIdx.x;
  float s = 0.0f;
  for (int w = 0; w < GRAM_BLOCKS; ++w) s += Tpart[(size_t)w * 256 + tid];
  Ts[tid] = s;
  __syncthreads();
  if (tid != 0) return;
  float u1[BLK], u2[BLK], t[BLK];
  for (int i = 0; i < BLK; ++i) u1[i] = 1.0f + 0.1f * hash11(i + 7u);
  for (int it = 0; it < 200; ++it) {
    float nrm = 0.0f;
    for (int i = 0; i < BLK; ++i) {
      float v = 0.0f;
      for (int k = 0; k < BLK; ++k) v += Ts[i * BLK + k] * u1[k];
      t[i] = v; nrm += v * v;
    }
    nrm = rsqrtf(fmaxf(nrm, 1e-30f));
    for (int i = 0; i < BLK; ++i) u1[i] = t[i] * nrm;
  }
  float lam1 = 0.0f;
  for (int i = 0; i < BLK; ++i) {
    float v = 0.0f;
    for (int k = 0; k < BLK; ++k) v += Ts[i * BLK + k] * u1[k];
    lam1 += u1[i] * v;
  }
  for (int i = 0; i < BLK; ++i) u2[i] = hash11(i * 31u + 3u);
  for (int it = 0; it < 200; ++it) {
    float d = 0.0f;
    for (int i = 0; i < BLK; ++i) d += u1[i] * u2[i];
    for (int i = 0; i < BLK; ++i) {
      float v = 0.0f;
      for (int k = 0; k < BLK; ++k) v += Ts[i * BLK + k] * u2[k];
      t[i] = v - lam1 * u1[i] * d;                   // deflation
    }
    float d2 = 0.0f, nrm = 0.0f;
    for (int i = 0; i < BLK; ++i) d2 += u1[i] * t[i];
    for (int i = 0; i < BLK; ++i) { t[i] -= d2 * u1[i]; nrm += t[i] * t[i]; }
    nrm = rsqrtf(fmaxf(nrm, 1e-30f));
    for (int i = 0; i < BLK; ++i) u2[i] = t[i] * nrm;
  }
  for (int i = 0; i < BLK; ++i) wv[i] = u2[i];
}

__global__ void k_top(const float* __restrict__ V, const float* __restrict__ wv,
                      float* __restrict__ top, int N) {
  int n = blockIdx.x * blockDim.x + threadIdx.x;
  if (n >= N) return;
  float s = 0.0f;
#pragma unroll
  for (int c = 0; c < BLK; ++c) s += V[n * BLK + c] * wv[c];
  top[n] = s;   // V columns orthonormal, wv unit => unit-norm eigenvector
}

// ---------------- GAT attention (segment softmax over dst) ----------------

__global__ void k_gatinit(const float* __restrict__ x, const float* __restrict__ lw,
                          const float* __restrict__ lb, float* __restrict__ h,
                          float* __restrict__ ssum, unsigned* __restrict__ key,
                          float* __restrict__ out, int N) {
  int i = blockIdx.x * blockDim.x + threadIdx.x;
  if (i >= N) return;
  h[i] = x[i] * lw[0] + lb[0];
  ssum[i] = 0.0f;
  key[i] = 0u;            // strictly below any finite fkey()
  out[i] = 0.0f;
}

__global__ void k_att1(const int* __restrict__ src, const int* __restrict__ dst,
                       const float* __restrict__ h, const float* __restrict__ top,
                       const float* __restrict__ aw, const float* __restrict__ ab,
                       const float* __restrict__ ow,
                       float* __restrict__ proj, unsigned* __restrict__ key, int E, int N) {
  int e = blockIdx.x * blockDim.x + threadIdx.x;
  if (e >= E + N) return;
  int s, t; float ea0, ea1;
  if (e < E) { s = src[e]; t = dst[e]; ea0 = top[s]; ea1 = top[t]; }
  else       { s = t = e - E; ea0 = 1.0f; ea1 = 1.0f; }     // self-loop, attr fill=1
  float xi = h[t], xj = h[s];
  float z0 = xi * aw[0] + xj * aw[2] + ab[0] + ea0;         // att_in_w is [in,out]
  float z1 = xi * aw[1] + xj * aw[3] + ab[1] + ea1;
  float t0 = z0 > 0.0f ? z0 : NEG_SLOPE * z0;
  float t1 = z1 > 0.0f ? z1 : NEG_SLOPE * z1;
  float p = t0 * ow[0] + t1 * ow[1];
  proj[e] = p;
  atomicMax(&key[t], fkey(p));
}

__global__ void k_att2(const int* __restrict__ dst, const float* __restrict__ proj,
                       const unsigned* __restrict__ key, float* __restrict__ expv,
                       float* __restrict__ ssum, int E, int N) {
  int e = blockIdx.x * blockDim.x + threadIdx.x;
  if (e >= E + N) return;
  int t = (e < E) ? dst[e] : (e - E);
  float ev = expf(proj[e] - fdec(key[t]));
  expv[e] = ev;
  atomicAdd(&ssum[t], ev);
}

__global__ void k_att3(const int* __restrict__ src, const int* __restrict__ dst,
                       const float* __restrict__ expv, const float* __restrict__ ssum,
                       const float* __restrict__ h, float* __restrict__ out, int E, int N) {
  int e = blockIdx.x * blockDim.x + threadIdx.x;
  if (e >= E + N) return;
  int s, t;
  if (e < E) { s = src[e]; t = dst[e]; } else { s = t = e - E; }
  float g = expv[e] / (ssum[t] + 1e-16f);
  atomicAdd(&out[t], g * h[s]);
}

// ---------------- host orchestration ----------------

static inline int gdiv(int n, int b) { return (n + b - 1) / b; }

extern "C" void kernel_launch(void* const* d_in, const int* in_sizes, int n_in,
                              void* d_out, int out_size, void* d_ws, size_t ws_size,
                              hipStream_t stream) {
  const float* x        = (const float*)d_in[0];
  const int*   ei       = (const int*)d_in[1];
  const float* lin_w    = (const float*)d_in[2];
  const float* lin_b    = (const float*)d_in[3];
  const float* att_in_w = (const float*)d_in[4];
  const float* att_in_b = (const float*)d_in[5];
  const float* att_out_w= (const float*)d_in[6];

  const int N = in_sizes[0];
  const int E = in_sizes[1] / 2;
  const int EN = E + N;
  const int* src = ei;
  const int* dst = ei + E;

  // workspace layout
  float* f     = (float*)d_ws;
  float* deg   = f;                 // N
  float* dinv  = f + (size_t)N;     // N
  float* mfl   = f + 2 * (size_t)N; // N
  float* h     = f + 3 * (size_t)N; // N
  float* top   = f + 4 * (size_t)N; // N
  float* ssum  = f + 5 * (size_t)N; // N
  unsigned* key = (unsigned*)(f + 6 * (size_t)N);          // N
  float* V     = f + 7 * (size_t)N;                        // N*16
  float* W     = f + 7 * (size_t)N + (size_t)N * BLK;      // N*16
  float* proj  = f + 7 * (size_t)N + 2 * (size_t)N * BLK;  // E+N
  float* expv  = proj + (size_t)EN;                        // E+N
  float* Gpart = expv + (size_t)EN;                        // GRAM_BLOCKS*256
  float* Cm    = Gpart + (size_t)GRAM_BLOCKS * 256;        // 256
  float* wv    = Cm + 256;                                 // 16
  int* rowstart = (int*)(wv + 16);                         // N+1
  int* cursor   = rowstart + (N + 1);                      // N
  int* csr_dst  = cursor + N;                              // E

  float* out = (float*)d_out;

  // --- degree / CSR build (one-time) ---
  k_zero<<<gdiv(N, 256), 256, 0, stream>>>(deg, N);
  k_deg<<<gdiv(E, 256), 256, 0, stream>>>(src, deg, E);
  k_dinv<<<gdiv(N, 256), 256, 0, stream>>>(deg, dinv, mfl, N);
  k_scan<<<1, 1, 0, stream>>>(deg, rowstart, cursor, N);
  k_fill<<<gdiv(E, 256), 256, 0, stream>>>(src, dst, rowstart, cursor, csr_dst, E);
  k_vinit<<<gdiv(N * BLK, 256), 256, 0, stream>>>(V, N * BLK);

  // --- block subspace iteration on B = 2I - L (WMMA orthonormalization) ---
  for (int it = 0; it < POWER_ITERS; ++it) {
    k_spmv<<<gdiv(N * BLK, 256), 256, 0, stream>>>(rowstart, csr_dst, dinv, mfl, V, W, N);
    k_gram<<<GRAM_BLOCKS, 32, 0, stream>>>(W, W, Gpart, N);
    k_cholinv<<<1, 256, 0, stream>>>(Gpart, Cm);
    k_apply<<<256, 32, 0, stream>>>(W, Cm, V, N);
  }

  // --- Rayleigh-Ritz: T = V^T B V, second Ritz vector -> top ---
  k_spmv<<<gdiv(N * BLK, 256), 256, 0, stream>>>(rowstart, csr_dst, dinv, mfl, V, W, N);
  k_gram<<<GRAM_BLOCKS, 32, 0, stream>>>(V, W, Gpart, N);
  k_smalleig<<<1, 256, 0, stream>>>(Gpart, wv);
  k_top<<<gdiv(N, 256), 256, 0, stream>>>(V, wv, top, N);

  // --- GAT attention with segment softmax over dst ---
  k_gatinit<<<gdiv(N, 256), 256, 0, stream>>>(x, lin_w, lin_b, h, ssum, key, out, N);
  k_att1<<<gdiv(EN, 256), 256, 0, stream>>>(src, dst, h, top, att_in_w, att_in_b,
                                            att_out_w, proj, key, E, N);
  k_att2<<<gdiv(EN, 256), 256, 0, stream>>>(dst, proj, key, expv, ssum, E, N);
  k_att3<<<gdiv(EN, 256), 256, 0, stream>>>(src, dst, expv, ssum, h, out, E, N);
}